// TripleViewNet_22479858827771
// MI455X (gfx1250) — compile-verified
//
#include <hip/hip_runtime.h>
#include <hip/hip_bf16.h>
#include <math.h>

// ---------------------------------------------------------------------------
// TripleViewNet (3x GAT branches + JK + pooling + MLP heads) for gfx1250.
// GEMMs use V_WMMA_F32_16X16X4_F32 (exact fp32). Edge aggregation is
// wave32-per-edge with L2-resident gathers/atomics (whole h table fits in
// the 192MB L2). BatchNorm fused into layer-0 GEMM A-path (templated so the
// hot K-loop is branch-free in both variants).
// ---------------------------------------------------------------------------

typedef __attribute__((ext_vector_type(2))) float v2f;
typedef __attribute__((ext_vector_type(8))) float v8f;

__device__ __forceinline__ unsigned fenc(float f) {
  unsigned u = __float_as_uint(f);
  return (u & 0x80000000u) ? ~u : (u | 0x80000000u);
}
__device__ __forceinline__ float fdec(unsigned u) {
  u = (u & 0x80000000u) ? (u & 0x7FFFFFFFu) : ~u;
  return __uint_as_float(u);
}

// ---------------- fills ----------------
__global__ void fill_f32(float* p, float v, int n) {
  int i = blockIdx.x * blockDim.x + threadIdx.x;
  if (i < n) p[i] = v;
}
__global__ void fill_u32(unsigned* p, unsigned v, int n) {
  int i = blockIdx.x * blockDim.x + threadIdx.x;
  if (i < n) p[i] = v;
}

// ---------------- BatchNorm stats (column mean / E[x^2]) ----------------
__global__ void bn_stats(const float* __restrict__ x, float* __restrict__ sum,
                         float* __restrict__ sq, int N, int F, int chunk) {
  int f = threadIdx.x;          // one thread per feature -> coalesced row reads
  int r0 = blockIdx.x * chunk;
  int r1 = r0 + chunk; if (r1 > N) r1 = N;
  float s = 0.f, q = 0.f;
  for (int r = r0; r < r1; ++r) {
    float v = x[(long)r * F + f];
    s += v; q += v * v;
  }
  atomicAdd(&sum[f], s);
  atomicAdd(&sq[f], q);
}
__global__ void bn_finalize(const float* sum, const float* sq, const float* g,
                            const float* b, float* scale, float* shift, int N, int F) {
  int f = threadIdx.x;
  if (f >= F) return;
  float mu = sum[f] / (float)N;
  float var = sq[f] / (float)N - mu * mu;
  float sc = g[f] * rsqrtf(var + 1e-5f);
  scale[f] = sc;
  shift[f] = b[f] - mu * sc;
}

// ---------------- WMMA fp32 GEMM: out[N x 64] = A[N x K] @ B[K x 64] -------
// Block = 256 threads = 8 waves = 2 (M) x 4 (N) tiles of 16x16 -> 32 rows/blk.
// BN=true fuses a = a*scale[k]+shift[k] on the A elements (compile-time, so
// the K-loop is straight-line: global_load -> ds_load -> v_wmma).
template <bool BN>
__global__ void gemm_wmma_f32(const float* __restrict__ A, int lda, int N, int K,
                              const float* __restrict__ B,
                              const float* __restrict__ scale,
                              const float* __restrict__ shift,
                              float* __restrict__ out) {
  extern __shared__ float sB[];              // K x 64 weight panel (<=32KB)
  for (int i = threadIdx.x; i < K * 64; i += blockDim.x) sB[i] = B[i];
  __syncthreads();

  const int wave = threadIdx.x >> 5;
  const int lane = threadIdx.x & 31;
  const int half = lane >> 4;                // 0: K pair {0,1}, 1: {2,3}
  const int l16  = lane & 15;
  const int waveM = wave & 1, waveN = wave >> 1;
  int row = blockIdx.x * 32 + waveM * 16 + l16;
  int rowc = row < N ? row : (N - 1);
  const int col = waveN * 16 + l16;
  const float* Arow = A + (long)rowc * lda;

  v8f acc = {};
  for (int k0 = 0; k0 < K; k0 += 4) {
    int ka = k0 + half * 2;
    float a0 = Arow[ka], a1 = Arow[ka + 1];
    if (BN) {
      a0 = a0 * scale[ka]     + shift[ka];
      a1 = a1 * scale[ka + 1] + shift[ka + 1];
    }
    v2f av = {a0, a1};
    v2f bv = {sB[ka * 64 + col], sB[(ka + 1) * 64 + col]};
    acc = __builtin_amdgcn_wmma_f32_16x16x4_f32(
        false, av, false, bv, (short)0, acc, false, false);
  }
  // D layout: VGPR r -> M = r + 8*half
  int rbase = blockIdx.x * 32 + waveM * 16 + half * 8;
#pragma unroll
  for (int r = 0; r < 8; ++r) {
    int orow = rbase + r;
    if (orow < N) out[(long)orow * 64 + col] = acc[r];
  }
}

// ---------------- attention coefficients: wave per node -------------------
__global__ void attn_coef(const float* __restrict__ h, const float* __restrict__ asrc,
                          const float* __restrict__ adst, float* __restrict__ as_,
                          float* __restrict__ ad_, int N) {
  int t = blockIdx.x * blockDim.x + threadIdx.x;
  int node = t >> 5, lane = t & 31;
  if (node >= N) return;
  float h0 = h[(long)node * 64 + lane];
  float h1 = h[(long)node * 64 + 32 + lane];
  float s0 = h0 * asrc[lane], s1 = h1 * asrc[32 + lane];
  float d0 = h0 * adst[lane], d1 = h1 * adst[32 + lane];
  for (int off = 16; off; off >>= 1) {
    s0 += __shfl_xor(s0, off, 32);
    s1 += __shfl_xor(s1, off, 32);
    d0 += __shfl_xor(d0, off, 32);
    d1 += __shfl_xor(d1, off, 32);
  }
  if (lane == 0) {
    as_[node * 2] = s0; as_[node * 2 + 1] = s1;
    ad_[node * 2] = d0; ad_[node * 2 + 1] = d1;
  }
}

// ---------------- edge pass 1: e = lrelu(as[src]+ad[dst]); seg-max --------
__global__ void edge_pass1(const int* __restrict__ ei, int E, int N,
                           const float* __restrict__ as_, const float* __restrict__ ad_,
                           float* __restrict__ exbuf, unsigned* __restrict__ menc) {
  int e = blockIdx.x * blockDim.x + threadIdx.x;
  if (e >= E + N) return;
  int src, dst;
  if (e < E) { src = ei[e]; dst = ei[E + e]; } else { src = dst = e - E; }
#pragma unroll
  for (int hh = 0; hh < 2; ++hh) {
    float v = as_[src * 2 + hh] + ad_[dst * 2 + hh];
    v = v >= 0.f ? v : 0.2f * v;             // leaky_relu(0.2)
    exbuf[(long)e * 2 + hh] = v;
    atomicMax(&menc[dst * 2 + hh], fenc(v));
  }
}

__global__ void m_decode(unsigned* m, int n) {
  int i = blockIdx.x * blockDim.x + threadIdx.x;
  if (i >= n) return;
  float f = fdec(m[i]);
  if (!isfinite(f)) f = 0.f;
  ((float*)m)[i] = f;
}

// ---------------- edge pass 2: ex = exp(e - m[dst]); seg-sum --------------
__global__ void edge_pass2(const int* __restrict__ ei, int E, int N,
                           float* __restrict__ exbuf, const float* __restrict__ mf,
                           float* __restrict__ den) {
  int e = blockIdx.x * blockDim.x + threadIdx.x;
  if (e >= E + N) return;
  int dst = (e < E) ? ei[E + e] : (e - E);
#pragma unroll
  for (int hh = 0; hh < 2; ++hh) {
    float ex = __expf(exbuf[(long)e * 2 + hh] - mf[dst * 2 + hh]);
    exbuf[(long)e * 2 + hh] = ex;
    atomicAdd(&den[dst * 2 + hh], ex);
  }
}

// ---------------- edge pass 3: out[dst] += h[src]*alpha (wave per edge) ---
__global__ void edge_pass3(const int* __restrict__ ei, int E, int N,
                           const float* __restrict__ h, const float* __restrict__ exbuf,
                           const float* __restrict__ den, float* __restrict__ outagg) {
  long t = (long)blockIdx.x * blockDim.x + threadIdx.x;
  int e = (int)(t >> 5), lane = (int)(t & 31);
  if (e >= E + N) return;
  int src, dst;
  if (e < E) { src = ei[e]; dst = ei[E + e]; } else { src = dst = e - E; }
  float a0 = exbuf[(long)e * 2]     / (den[dst * 2]     + 1e-16f);
  float a1 = exbuf[(long)e * 2 + 1] / (den[dst * 2 + 1] + 1e-16f);
  atomicAdd(&outagg[(long)dst * 64 + lane],      h[(long)src * 64 + lane] * a0);
  atomicAdd(&outagg[(long)dst * 64 + 32 + lane], h[(long)src * 64 + 32 + lane] * a1);
}

// ---------------- epilogue: +bias, ELU, write into JK column block --------
__global__ void epilogue_elu(const float* __restrict__ agg, const float* __restrict__ bias,
                             float* __restrict__ jkcol, int N) {
  int i = blockIdx.x * blockDim.x + threadIdx.x;
  if (i >= N * 64) return;
  int node = i >> 6, ch = i & 63;
  float v = agg[i] + bias[ch];
  v = v > 0.f ? v : (__expf(v) - 1.f);       // elu
  jkcol[(long)node * 192 + ch] = v;
}

// ---------------- per-node LayerNorm + segment max/sum pooling ------------
__global__ void ln_pool(const float* __restrict__ jk, const float* __restrict__ g_,
                        const float* __restrict__ b_, const int* __restrict__ batch,
                        unsigned* __restrict__ gmax, float* __restrict__ gsum, int N) {
  int t = blockIdx.x * blockDim.x + threadIdx.x;
  int node = t >> 5, lane = t & 31;
  if (node >= N) return;
  float v[6]; float s = 0.f, q = 0.f;
#pragma unroll
  for (int k = 0; k < 6; ++k) {
    v[k] = jk[(long)node * 192 + k * 32 + lane];
    s += v[k]; q += v[k] * v[k];
  }
  for (int off = 16; off; off >>= 1) {
    s += __shfl_xor(s, off, 32);
    q += __shfl_xor(q, off, 32);
  }
  float mu = s * (1.f / 192.f);
  float var = q * (1.f / 192.f) - mu * mu;
  float rs = rsqrtf(var + 1e-5f);
  int grp = batch[node];
#pragma unroll
  for (int k = 0; k < 6; ++k) {
    int idx = k * 32 + lane;
    float y = (v[k] - mu) * rs * g_[idx] + b_[idx];
    atomicMax(&gmax[grp * 192 + idx], fenc(y));
    atomicAdd(&gsum[grp * 192 + idx], y);
  }
}

__global__ void count_nodes(const int* __restrict__ batch, float* cnt, int N) {
  int i = blockIdx.x * blockDim.x + threadIdx.x;
  if (i < N) atomicAdd(&cnt[batch[i]], 1.f);
}

__global__ void pool_final(const unsigned* __restrict__ gmax, const float* __restrict__ gsum,
                           const float* __restrict__ cnt, float* __restrict__ henc) {
  int g = blockIdx.x, j = threadIdx.x;       // 192 threads
  float mx = fdec(gmax[g * 192 + j]);
  if (!isfinite(mx)) mx = 0.f;
  float mean = gsum[g * 192 + j] / fmaxf(cnt[g], 1.f);
  henc[(long)g * 384 + j] = mx;
  henc[(long)g * 384 + 192 + j] = mean;
}

// ---------------- heads -----------------------------------------------------
__global__ void proj_kernel(const float* __restrict__ henc, const float* __restrict__ W,
                            const float* __restrict__ b, float* __restrict__ pout, int G) {
  int g = blockIdx.x, br = blockIdx.y, t = threadIdx.x;   // 128 threads
  const float* hr = henc + ((long)br * G + g) * 384;
  float acc = b[t];
  for (int k = 0; k < 384; ++k) acc += hr[k] * W[k * 128 + t];
  pout[((long)br * G + g) * 128 + t] = fmaxf(acc, 0.f);
}

__global__ void head_kernel(const float* __restrict__ p, const float* gW1, const float* gb1,
                            const float* gW2, const float* gb2, const float* cW1,
                            const float* cb1, const float* cW2, const float* cb2,
                            const float* cW3, const float* cb3, float* __restrict__ out,
                            int G) {
  __shared__ float savg[128], sz1[64], sw[3], sf[128], sz[128], sz2[64];
  int g = blockIdx.x, t = threadIdx.x;       // 128 threads
  float p0 = p[((long)0 * G + g) * 128 + t];
  float p1 = p[((long)1 * G + g) * 128 + t];
  float p2 = p[((long)2 * G + g) * 128 + t];
  savg[t] = (p0 + p1 + p2) * (1.f / 3.f);
  __syncthreads();
  if (t < 64) {
    float a = gb1[t];
    for (int k = 0; k < 128; ++k) a += savg[k] * gW1[k * 64 + t];
    sz1[t] = fmaxf(a, 0.f);
  }
  __syncthreads();
  if (t == 0) {
    float l[3];
    for (int v = 0; v < 3; ++v) {
      float a = gb2[v];
      for (int k = 0; k < 64; ++k) a += sz1[k] * gW2[k * 3 + v];
      l[v] = a;
    }
    float mx = fmaxf(l[0], fmaxf(l[1], l[2]));
    float e0 = __expf(l[0] - mx), e1 = __expf(l[1] - mx), e2 = __expf(l[2] - mx);
    float inv = 1.f / (e0 + e1 + e2);
    sw[0] = e0 * inv; sw[1] = e1 * inv; sw[2] = e2 * inv;
  }
  __syncthreads();
  sf[t] = sw[0] * p0 + sw[1] * p1 + sw[2] * p2;
  __syncthreads();
  {
    float a = cb1[t];
    for (int k = 0; k < 128; ++k) a += sf[k] * cW1[k * 128 + t];
    sz[t] = fmaxf(a, 0.f);
  }
  __syncthreads();
  if (t < 64) {
    float a = cb2[t];
    for (int k = 0; k < 128; ++k) a += sz[k] * cW2[k * 64 + t];
    sz2[t] = fmaxf(a, 0.f);
  }
  __syncthreads();
  if (t == 0) {
    float a = cb3[0];
    for (int k = 0; k < 64; ++k) a += sz2[k] * cW3[k];
    out[g] = a;
  }
}

// ---------------------------------------------------------------------------
extern "C" void kernel_launch(void* const* d_in, const int* in_sizes, int n_in,
                              void* d_out, int out_size, void* d_ws, size_t ws_size,
                              hipStream_t stream) {
  const int N = in_sizes[4];                 // batch length
  const int F = in_sizes[0] / N;             // 128
  const int E = in_sizes[1] / 2;             // edges per view
  const int G = 256;
  const int EN = E + N;

  const float* x     = (const float*)d_in[0];
  const int*   batch = (const int*)d_in[4];
  const float* bn_g  = (const float*)d_in[5];
  const float* bn_b  = (const float*)d_in[6];
  const float* jk_g  = (const float*)d_in[43];
  const float* jk_b  = (const float*)d_in[44];
  const float* projW = (const float*)d_in[45];
  const float* projB = (const float*)d_in[46];

  // ---- workspace carve-out (floats) ----
  float* w = (float*)d_ws;
  size_t off = 0;
  float*    jk      = w + off; off += (size_t)N * 192;
  float*    h_lin   = w + off; off += (size_t)N * 64;
  float*    out_agg = w + off; off += (size_t)N * 64;
  float*    a_s     = w + off; off += (size_t)N * 2;
  float*    a_d     = w + off; off += (size_t)N * 2;
  float*    mbuf    = w + off; off += (size_t)N * 2;     // uint-encoded then float
  float*    den     = w + off; off += (size_t)N * 2;
  float*    exbuf   = w + off; off += (size_t)EN * 2;
  float*    bn_sum  = w + off; off += F;
  float*    bn_sq   = w + off; off += F;
  float*    bn_sc   = w + off; off += F;
  float*    bn_sh   = w + off; off += F;
  float*    cntf    = w + off; off += G;
  unsigned* gmax    = (unsigned*)(w + off); off += (size_t)G * 192;
  float*    gsum    = w + off; off += (size_t)G * 192;
  float*    henc    = w + off; off += (size_t)3 * G * 384;
  float*    pbr     = w + off; off += (size_t)3 * G * 128;
  (void)ws_size; (void)n_in;

  const int TB = 256;
  auto blocks = [](long n, int tb) { return (int)((n + tb - 1) / tb); };

  // ---- BatchNorm (training-mode stats) -> fused scale/shift ----
  fill_f32<<<blocks(2 * F, TB), TB, 0, stream>>>(bn_sum, 0.f, 2 * F); // sum+sq adjacent
  int chunk = (N + 255) / 256;
  bn_stats<<<256, F, 0, stream>>>(x, bn_sum, bn_sq, N, F, chunk);
  bn_finalize<<<1, F, 0, stream>>>(bn_sum, bn_sq, bn_g, bn_b, bn_sc, bn_sh, N, F);

  // ---- per-graph node counts (same batch for every branch) ----
  fill_f32<<<blocks(G, TB), TB, 0, stream>>>(cntf, 0.f, G);
  count_nodes<<<blocks(N, TB), TB, 0, stream>>>(batch, cntf, N);

  for (int br = 0; br < 3; ++br) {
    const int* ei = (const int*)d_in[1 + br];
    for (int l = 0; l < 3; ++l) {
      int base = 7 + br * 12 + l * 4;
      const float* W    = (const float*)d_in[base + 0];
      const float* aSrc = (const float*)d_in[base + 1];
      const float* aDst = (const float*)d_in[base + 2];
      const float* bias = (const float*)d_in[base + 3];
      int K = (l == 0) ? F : 64;
      const float* A = (l == 0) ? x : (jk + (l - 1) * 64);
      int lda = (l == 0) ? F : 192;

      // h = A @ W (WMMA fp32, BN fused on layer 0; branch-free inner loops)
      if (l == 0) {
        gemm_wmma_f32<true><<<blocks((long)N, 32), TB,
                              (size_t)K * 64 * sizeof(float), stream>>>(
            A, lda, N, K, W, bn_sc, bn_sh, h_lin);
      } else {
        gemm_wmma_f32<false><<<blocks((long)N, 32), TB,
                               (size_t)K * 64 * sizeof(float), stream>>>(
            A, lda, N, K, W, nullptr, nullptr, h_lin);
      }

      // attention source/dest coefficients
      attn_coef<<<blocks((long)N * 32, TB), TB, 0, stream>>>(h_lin, aSrc, aDst, a_s, a_d, N);

      // segment softmax over incoming edges (incl. self-loops)
      fill_u32<<<blocks(N * 2, TB), TB, 0, stream>>>((unsigned*)mbuf, 0u, N * 2);
      fill_f32<<<blocks(N * 2, TB), TB, 0, stream>>>(den, 0.f, N * 2);
      fill_f32<<<blocks((long)N * 64, TB), TB, 0, stream>>>(out_agg, 0.f, N * 64);
      edge_pass1<<<blocks(EN, TB), TB, 0, stream>>>(ei, E, N, a_s, a_d, exbuf, (unsigned*)mbuf);
      m_decode<<<blocks(N * 2, TB), TB, 0, stream>>>((unsigned*)mbuf, N * 2);
      edge_pass2<<<blocks(EN, TB), TB, 0, stream>>>(ei, E, N, exbuf, mbuf, den);
      edge_pass3<<<blocks((long)EN * 32, TB), TB, 0, stream>>>(ei, E, N, h_lin, exbuf, den, out_agg);

      // bias + ELU -> JK column block l (also the next layer's input)
      epilogue_elu<<<blocks((long)N * 64, TB), TB, 0, stream>>>(out_agg, bias, jk + l * 64, N);
    }

    // LayerNorm over 192-dim JK features + per-graph max/mean pooling
    fill_u32<<<blocks(G * 192, TB), TB, 0, stream>>>(gmax, 0u, G * 192);
    fill_f32<<<blocks(G * 192, TB), TB, 0, stream>>>(gsum, 0.f, G * 192);
    ln_pool<<<blocks((long)N * 32, TB), TB, 0, stream>>>(jk, jk_g, jk_b, batch, gmax, gsum, N);
    pool_final<<<G, 192, 0, stream>>>(gmax, gsum, cntf, henc + (size_t)br * G * 384);
  }

  // ---- heads: projection per branch, gated fusion, classifier ----
  dim3 pg(G, 3);
  proj_kernel<<<pg, 128, 0, stream>>>(henc, projW, projB, pbr, G);
  head_kernel<<<G, 128, 0, stream>>>(
      pbr,
      (const float*)d_in[47], (const float*)d_in[48],   // gate_W1, gate_b1
      (const float*)d_in[49], (const float*)d_in[50],   // gate_W2, gate_b2
      (const float*)d_in[51], (const float*)d_in[52],   // cls_W1, cls_b1
      (const float*)d_in[53], (const float*)d_in[54],   // cls_W2, cls_b2
      (const float*)d_in[55], (const float*)d_in[56],   // cls_W3, cls_b3
      (float*)d_out, G);
  (void)out_size;
}